// ReplaceBackgroundOperation_42580305773206
// MI455X (gfx1250) — compile-verified
//
#include <hip/hip_runtime.h>

typedef float v2f __attribute__((ext_vector_type(2)));
typedef float v4f __attribute__((ext_vector_type(4)));
typedef float v8f __attribute__((ext_vector_type(8)));

#define B_      32
#define C_      10
#define HW_     (512 * 512)        // 262144 floats per plane
#define HW4_    (HW_ / 4)          // 65536 v4f per plane
#define SPLIT_  8                  // blocks per (b,c) plane in the reduce
#define CHUNK4_ (HW4_ / SPLIT_)    // 8192 v4f per reduce block
#define T1_     256

// ---------------------------------------------------------------------------
// Pass 1: per-(b,c) channel sums. WMMA f32 16x16x4 used as a streaming
// accumulator: B = 1/16 everywhere, so sum(all D VGPRs over all lanes) equals
// the exact sum of every element fed through A (element->lane mapping is
// irrelevant because we reduce the whole accumulator at the end).
// ---------------------------------------------------------------------------
__global__ __launch_bounds__(T1_)
void sum_kernel(const float* __restrict__ grid, float* __restrict__ partial) {
    const int plane = blockIdx.x / SPLIT_;   // b*C + c
    const int sub   = blockIdx.x % SPLIT_;
    const v4f* __restrict__ p =
        (const v4f*)grid + (size_t)plane * HW4_ + (size_t)sub * CHUNK4_;

    v8f acc = {};
    v2f bfrag;
    bfrag[0] = 0.0625f;   // 1/16 in every lane -> all-(1/16) 4x16 B matrix
    bfrag[1] = 0.0625f;

    const int t = threadIdx.x;
    #pragma unroll 4
    for (int i = 0; i < CHUNK4_ / T1_; ++i) {        // 32 iterations
        v4f v = __builtin_nontemporal_load(&p[i * T1_ + t]);
        v2f a0; a0[0] = v[0]; a0[1] = v[1];
        v2f a1; a1[0] = v[2]; a1[1] = v[3];
        acc = __builtin_amdgcn_wmma_f32_16x16x4_f32(
                  false, a0, false, bfrag, (short)0, acc, false, false);
        acc = __builtin_amdgcn_wmma_f32_16x16x4_f32(
                  false, a1, false, bfrag, (short)0, acc, false, false);
    }

    float s = acc[0] + acc[1] + acc[2] + acc[3] +
              acc[4] + acc[5] + acc[6] + acc[7];

    __shared__ float red[T1_];
    red[t] = s;
    __syncthreads();
    for (int w = T1_ / 2; w > 0; w >>= 1) {
        if (t < w) red[t] += red[t + w];
        __syncthreads();
    }
    if (t == 0) partial[blockIdx.x] = red[0];
}

// ---------------------------------------------------------------------------
// Pass 2: per-batch argmax over C channel sums (first-max, matching argmax).
// ---------------------------------------------------------------------------
__global__ void argmax_kernel(const float* __restrict__ partial,
                              int* __restrict__ bgIdx) {
    const int b = threadIdx.x;
    if (b >= B_) return;
    float best = -3.402823466e38f;
    int bi = 0;
    for (int c = 0; c < C_; ++c) {
        float s = 0.0f;
        for (int k = 0; k < SPLIT_; ++k)
            s += partial[(b * C_ + c) * SPLIT_ + k];
        if (s > best) { best = s; bi = c; }
    }
    bgIdx[b] = bi;
}

// ---------------------------------------------------------------------------
// Pass 3: masked rewrite. One thread per v4f spatial position per batch.
// Each 256-thread block covers a contiguous 256-v4f span of one batch, so
// bg/target branches are wave-uniform. Zero bg channel at mask, then set
// target channel at mask (same order as reference: target overrides bg).
// ---------------------------------------------------------------------------
__global__ __launch_bounds__(T1_)
void replace_kernel(const float* __restrict__ grid,
                    const int* __restrict__ bgIdx,
                    const int* __restrict__ tptr,
                    float* __restrict__ out) {
    const int g   = blockIdx.x * blockDim.x + threadIdx.x;
    const int b   = g >> 16;          // HW4_ == 65536 positions per batch
    const int pos = g & (HW4_ - 1);
    const int bg  = bgIdx[b];
    const int tgt = *tptr;

    const v4f* __restrict__ in4 = (const v4f*)grid;
    v4f* __restrict__ out4      = (v4f*)out;
    const size_t bbase = (size_t)b * C_ * HW4_;

    const v4f bgv = in4[bbase + (size_t)bg * HW4_ + pos];
    const bool m0 = bgv[0] > 0.5f, m1 = bgv[1] > 0.5f,
               m2 = bgv[2] > 0.5f, m3 = bgv[3] > 0.5f;

    #pragma unroll
    for (int c = 0; c < C_; ++c) {
        const size_t idx = bbase + (size_t)c * HW4_ + pos;
        v4f v = __builtin_nontemporal_load(&in4[idx]);
        if (c == bg) {
            if (m0) v[0] = 0.0f;
            if (m1) v[1] = 0.0f;
            if (m2) v[2] = 0.0f;
            if (m3) v[3] = 0.0f;
        }
        if (c == tgt) {
            if (m0) v[0] = 1.0f;
            if (m1) v[1] = 1.0f;
            if (m2) v[2] = 1.0f;
            if (m3) v[3] = 1.0f;
        }
        __builtin_nontemporal_store(v, &out4[idx]);
    }
}

// ---------------------------------------------------------------------------
extern "C" void kernel_launch(void* const* d_in, const int* in_sizes, int n_in,
                              void* d_out, int out_size, void* d_ws, size_t ws_size,
                              hipStream_t stream) {
    const float* grid = (const float*)d_in[0];
    const int* target = (const int*)d_in[1];
    float* out        = (float*)d_out;

    float* partial = (float*)d_ws;                                  // B*C*SPLIT floats
    int* bgIdx     = (int*)((char*)d_ws +
                            (size_t)B_ * C_ * SPLIT_ * sizeof(float)); // B ints

    sum_kernel<<<B_ * C_ * SPLIT_, T1_, 0, stream>>>(grid, partial);
    argmax_kernel<<<1, 32, 0, stream>>>(partial, bgIdx);
    replace_kernel<<<(B_ * HW4_) / T1_, T1_, 0, stream>>>(grid, bgIdx, target, out);
}